// Convolution_layer_33097017983241
// MI455X (gfx1250) — compile-verified
//
#include <hip/hip_runtime.h>

typedef float v2f __attribute__((ext_vector_type(2)));
typedef float v4f __attribute__((ext_vector_type(4)));
typedef float v8f __attribute__((ext_vector_type(8)));

#define NETSUM 512
#define KD 512

// ---------------------------------------------------------------------------
// Kernel A: Wt[a,b] = W[b,a]  (makes weight access coalesced in compute_d)
// ---------------------------------------------------------------------------
__global__ void __launch_bounds__(1024) transpose_w(const float* __restrict__ W,
                                                    float* __restrict__ Wt) {
    __shared__ float tile[32][33];
    const int bx = blockIdx.x * 32, by = blockIdx.y * 32;
    const int tx = threadIdx.x, ty = threadIdx.y;
    tile[ty][tx] = W[(size_t)(by + ty) * KD + bx + tx];
    __syncthreads();
    Wt[(size_t)(bx + ty) * KD + by + tx] = tile[tx][ty];
}

// ---------------------------------------------------------------------------
// Kernel B: one block per image i.  P[a,b] = X[i,a,b]*Wt[a,b].
//   d[i,t] = rowsum(P)[t] + colsum(P)[t]
// 512 threads: lane owns columns 4*cg..4*cg+3 (cg = tid&127) within row group
// rg = tid>>7 (rows rg*128..rg*128+127).  Column sums accumulate in registers;
// row sums via wave32 xor-butterfly (result uniform across lanes), and lane k
// latches row a0+32*j+k into a register -- no divergent stores in the loop.
// Deterministic LDS partial combine (no float atomics).
// ---------------------------------------------------------------------------
__global__ void __launch_bounds__(512) compute_d(const float* __restrict__ X,
                                                 const float* __restrict__ Wt,
                                                 float* __restrict__ d) {
    __shared__ float rowpart[KD * 4];   // [row][wave-in-rowgroup]
    __shared__ float colpart[KD * 4];   // [col][rowgroup]

    const int tid  = threadIdx.x;
    const int i    = blockIdx.x;
    const int cg   = tid & 127;
    const int rg   = tid >> 7;
    const int c4   = cg << 2;
    const int lane = tid & 31;
    const int wir  = (tid >> 5) & 3;    // wave index within row group

    const float* xbase = X + (size_t)i * KD * KD + c4;
    const float* wbase = Wt + c4;

    v4f cacc = {0.f, 0.f, 0.f, 0.f};
    const int a0 = rg * 128;

    float racc0 = 0.f, racc1 = 0.f, racc2 = 0.f, racc3 = 0.f;
    #pragma unroll
    for (int j = 0; j < 4; ++j) {
        float rj = 0.f;
        #pragma unroll 4
        for (int k = 0; k < 32; ++k) {
            const int a = a0 + j * 32 + k;
            v4f x = __builtin_nontemporal_load((const v4f*)(xbase + (size_t)a * KD));
            v4f w = *(const v4f*)(wbase + (size_t)a * KD);
            v4f p = x * w;
            cacc += p;
            float rp = (p[0] + p[1]) + (p[2] + p[3]);
            rp += __shfl_xor(rp, 16, 32);
            rp += __shfl_xor(rp,  8, 32);
            rp += __shfl_xor(rp,  4, 32);
            rp += __shfl_xor(rp,  2, 32);
            rp += __shfl_xor(rp,  1, 32);
            // butterfly leaves the full row sum in every lane; lane k keeps row a
            rj = (lane == k) ? rp : rj;
        }
        if (j == 0) racc0 = rj;
        if (j == 1) racc1 = rj;
        if (j == 2) racc2 = rj;
        if (j == 3) racc3 = rj;
    }
    // wave writes its 32 row sums (rows a0+32*j+lane), unconditionally
    rowpart[(a0 +  0 + lane) * 4 + wir] = racc0;
    rowpart[(a0 + 32 + lane) * 4 + wir] = racc1;
    rowpart[(a0 + 64 + lane) * 4 + wir] = racc2;
    rowpart[(a0 + 96 + lane) * 4 + wir] = racc3;

    colpart[(c4 + 0) * 4 + rg] = cacc[0];
    colpart[(c4 + 1) * 4 + rg] = cacc[1];
    colpart[(c4 + 2) * 4 + rg] = cacc[2];
    colpart[(c4 + 3) * 4 + rg] = cacc[3];
    __syncthreads();

    float rs = (rowpart[tid*4+0] + rowpart[tid*4+1]) + (rowpart[tid*4+2] + rowpart[tid*4+3]);
    float cs = (colpart[tid*4+0] + colpart[tid*4+1]) + (colpart[tid*4+2] + colpart[tid*4+3]);
    d[(size_t)i * KD + tid] = rs + cs;
}

// ---------------------------------------------------------------------------
// Kernel C: out[i,r,c] = relu(0.1*(d[i,r]+d[i,c])) via V_WMMA_F32_16X16X4_F32.
//   D = A x B with A = [0.1*d_r | 1 | 0 | 0] (16x4), B = [1; 0.1*d_c; *; *].
// LDS loads are unconditional (lanes 0-15 and 16-31 fetch the same 16 values);
// operand registers are built with cndmask selects so EXEC stays all-1s and
// there is exactly one ds_load per tile.  B is filled so both plausible f32
// B half-lane layouts give the same product (ambiguous rows hit zero A cols).
// ---------------------------------------------------------------------------
__global__ void __launch_bounds__(256) outer_relu(const float* __restrict__ d,
                                                  float* __restrict__ out) {
    __shared__ float dsh[KD];
    const int i   = blockIdx.x;
    const int tid = threadIdx.x;
    dsh[tid]       = 0.1f * d[(size_t)i * KD + tid];
    dsh[tid + 256] = 0.1f * d[(size_t)i * KD + tid + 256];
    __syncthreads();

    const int wave = tid >> 5;
    const int lane = tid & 31;
    const int r0   = (blockIdx.y * 8 + wave) << 4;   // 16-row stripe
    const bool lo  = lane < 16;
    const int l15  = lane & 15;

    // A 16x4 (documented layout): VGPR0 = {K=0 | K=2}, VGPR1 = {K=1 | K=3}
    const float dr = dsh[r0 + l15];      // unconditional, all lanes
    v2f A;
    A[0] = lo ? dr  : 0.f;               // col0 = 0.1*d_r ; col2 = 0
    A[1] = lo ? 1.f : 0.f;               // col1 = 1       ; col3 = 0

    float* obase = out + ((size_t)i * KD + (size_t)(r0 + (lo ? 0 : 8))) * KD + l15;

    for (int c0 = 0; c0 < KD; c0 += 16) {
        const float dc = dsh[c0 + l15];  // unconditional, one ds_load per tile
        v2f B;
        B[0] = lo ? 1.f : dc;            // row0 = 1       (both layouts)
        B[1] = lo ? dc  : 0.f;           // row1 = 0.1*d_c (both layouts)
        v8f acc = {};
        acc = __builtin_amdgcn_wmma_f32_16x16x4_f32(
            /*neg_a=*/false, A, /*neg_b=*/false, B,
            /*c_mod=*/(short)0, acc, /*reuse_a=*/false, /*reuse_b=*/false);
        #pragma unroll
        for (int v = 0; v < 8; ++v) {
            float val = fmaxf(acc[v], 0.f);
            __builtin_nontemporal_store(val, obase + (size_t)v * KD + c0);
        }
    }
}

// ---------------------------------------------------------------------------
extern "C" void kernel_launch(void* const* d_in, const int* in_sizes, int n_in,
                              void* d_out, int out_size, void* d_ws, size_t ws_size,
                              hipStream_t stream) {
    const float* X = (const float*)d_in[0];   // [512,512,512] f32
    const float* W = (const float*)d_in[1];   // [512,512] f32
    float* out  = (float*)d_out;              // [512,512,512] f32
    float* Wt   = (float*)d_ws;               // 1 MB
    float* dbuf = Wt + (size_t)KD * KD;       // 1 MB

    transpose_w<<<dim3(KD / 32, KD / 32), dim3(32, 32), 0, stream>>>(W, Wt);
    compute_d  <<<dim3(NETSUM),           dim3(512),    0, stream>>>(X, Wt, dbuf);
    outer_relu <<<dim3(NETSUM, 4),        dim3(256),    0, stream>>>(dbuf, out);
}